// ClofNet_31825707663892
// MI455X (gfx1250) — compile-verified
//
#include <hip/hip_runtime.h>

#define N_NODES 16384
#define N_EDGES 262144
#define HID 128
#define NUM_RBF 12
#define NLAYERS 4

typedef __attribute__((ext_vector_type(2))) float v2f;
typedef __attribute__((ext_vector_type(8))) float v8f;

// RBF constants: means = linspace(exp(-5), 1, 12); beta = (2/12*(1-exp(-5)))^-2
#define RBF_M0         0.00673794700f
#define RBF_DM         0.09029655027f
#define RBF_BETA       36.4900779f
#define PI_OVER_CUTOFF 0.62831853072f
#define CUTOFF         5.0f

__device__ __forceinline__ float silu_f(float x) { return x / (1.0f + expf(-x)); }

// ---------------------------------------------------------------------------
// Node prep: s = node_emb[z]; lnn = LayerNorm(neigh_emb[z]); zero v buffer.
// One block (128 threads) per node row.
// ---------------------------------------------------------------------------
__global__ __launch_bounds__(128) void node_prep_kernel(
    const int* __restrict__ z, const float* __restrict__ node_emb,
    const float* __restrict__ neigh_emb, float* __restrict__ s,
    float* __restrict__ lnn, float* __restrict__ v)
{
    __shared__ float red[128];
    const int row = blockIdx.x;
    const int t = threadIdx.x;
    const int zi = z[row];

    float ne = neigh_emb[(size_t)zi * HID + t];

    red[t] = ne; __syncthreads();
    #pragma unroll
    for (int o = 64; o > 0; o >>= 1) { if (t < o) red[t] += red[t + o]; __syncthreads(); }
    float m = red[0] * (1.0f / HID);
    __syncthreads();

    float dv = ne - m;
    red[t] = dv * dv; __syncthreads();
    #pragma unroll
    for (int o = 64; o > 0; o >>= 1) { if (t < o) red[t] += red[t + o]; __syncthreads(); }
    float rstd = rsqrtf(red[0] * (1.0f / HID) + 1e-5f);

    lnn[(size_t)row * HID + t] = dv * rstd;
    s[(size_t)row * HID + t]   = node_emb[(size_t)zi * HID + t];

    size_t vb = (size_t)row * 3 * HID + t;
    v[vb] = 0.0f; v[vb + HID] = 0.0f; v[vb + 2 * HID] = 0.0f;
}

// ---------------------------------------------------------------------------
// Node GEMM: out = SiLU(LayerNorm(X @ W + bias)).  X:[N,128] W:[128,128]
// 128 threads = 4 waves; each wave computes a 16x128 slice via 8 WMMA C-tiles
// with K=128 (32 x v_wmma_f32_16x16x4_f32).
// W is staged through LDS in two 32KB halves, SWIZZLED so that the WMMA
// B-fragment {W[k][col], W[k+1][col]} is one aligned 8-byte LDS load:
//   element (k, col)  ->  Sm[ (k>>1)*256 + col*2 + (k&1) ]
// The same 32KB region is then reused to stage C for the LayerNorm epilogue.
// ---------------------------------------------------------------------------
__global__ __launch_bounds__(128) void gemm_ln_silu_kernel(
    const float* __restrict__ X, const float* __restrict__ W,
    const float* __restrict__ bias, float* __restrict__ out)
{
    __shared__ float Sm[64 * HID];        // 32KB: W-half staging, then C staging
    __shared__ float mean_s[64];
    __shared__ float rstd_s[64];

    const int tid = threadIdx.x;
    const int wave = tid >> 5;
    const int lane = tid & 31;
    const int laneLo = lane & 15;
    const int laneHi = lane >> 4;
    const int rowblock = blockIdx.x * 64;
    const int rowbase = rowblock + wave * 16;

    v8f acc[8];
    #pragma unroll
    for (int n = 0; n < 8; ++n) { v8f zz = {0,0,0,0,0,0,0,0}; acc[n] = zz; }

    const float* xrow = X + (size_t)(rowbase + laneLo) * HID;

    for (int h = 0; h < 2; ++h) {
        // stage 64 K-rows of W into LDS with the (k,k+1)-pair swizzle
        const float4* Wg = (const float4*)(W + (size_t)h * 64 * HID);
        #pragma unroll
        for (int t = 0; t < (64 * HID / 4) / 128; ++t) {
            const int f4 = tid + t * 128;        // float4 index in row-major half
            const int k = f4 >> 5;               // K row 0..63
            const int c0 = (f4 & 31) * 4;        // starting col
            const float4 val = Wg[f4];
            const int base = (k >> 1) * 256 + c0 * 2 + (k & 1);
            Sm[base + 0] = val.x;
            Sm[base + 2] = val.y;
            Sm[base + 4] = val.z;
            Sm[base + 6] = val.w;
        }
        __syncthreads();

        const float* xh = xrow + h * 64;
        const v2f* Bp = (const v2f*)Sm;          // 8-byte fragment elements
        #pragma unroll 4
        for (int t = 0; t < 16; ++t) {
            const int k0 = 4 * t + 2 * laneHi;   // A layout: lanes 16-31 hold K+2
            v2f a;
            a.x = xh[k0];
            a.y = xh[k0 + 1];
            const int prow = (2 * t + laneHi) * 128;   // (k0>>1)*128
            #pragma unroll
            for (int n = 0; n < 8; ++n) {
                const v2f b = Bp[prow + n * 16 + laneLo];
                acc[n] = __builtin_amdgcn_wmma_f32_16x16x4_f32(
                    false, a, false, b, (short)0, acc[n], false, false);
            }
        }
        __syncthreads();   // all waves done with this W half before reuse
    }

    // C layout: VGPR r, lane<16 -> (M=r, N=lane); lane>=16 -> (M=r+8, N=lane-16)
    #pragma unroll
    for (int n = 0; n < 8; ++n) {
        const int col = n * 16 + laneLo;
        const float bc = bias[col];
        #pragma unroll
        for (int r = 0; r < 8; ++r) {
            const int M = r + 8 * laneHi;
            Sm[(wave * 16 + M) * HID + col] = acc[n][r] + bc;
        }
    }
    __syncthreads();

    if (tid < 64) {
        const float* crow = Sm + tid * HID;
        float sm = 0.0f;
        #pragma unroll 8
        for (int c = 0; c < HID; ++c) sm += crow[c];
        const float m = sm * (1.0f / HID);
        float vs = 0.0f;
        #pragma unroll 8
        for (int c = 0; c < HID; ++c) { float d = crow[c] - m; vs += d * d; }
        mean_s[tid] = m;
        rstd_s[tid] = rsqrtf(vs * (1.0f / HID) + 1e-5f);
    }
    __syncthreads();

    #pragma unroll 4
    for (int i = 0; i < 64; ++i) {
        const int e = tid + i * 128;
        const int r = e >> 7, c = e & 127;
        const float x = (Sm[e] - mean_s[r]) * rstd_s[r];
        out[(size_t)(rowblock + r) * HID + c] = silu_f(x);
    }
}

// ---------------------------------------------------------------------------
// Edge pass. Each wave owns 16 edges; recomputes the 12 RBF values from pos
// and does rbf[16x12] @ Wsm[12x128] via 3 K-steps of v_wmma_f32_16x16x4_f32.
// W staged in LDS with the same pair-swizzle (one ds_load_b64 per fragment).
// MODE 0: radial = SiLU(rbf@Wr+br);  s[i] += radial * lnn[j]        (atomic)
// MODE 1: radial = SiLU(rbf@Wr+br);  v[i][d] += radial*unit_d*s1[j] (atomic)
// MODE 2: ep = rbf@We[l];            s[i] += (val[j]+ep)*bounds     (atomic)
// ---------------------------------------------------------------------------
template<int MODE>
__global__ __launch_bounds__(256) void edge_pass_kernel(
    const float* __restrict__ pos,
    const int* __restrict__ jsrc, const int* __restrict__ idst,
    const float* __restrict__ Wsm,   // [12,128]
    const float* __restrict__ bias,  // [128] (MODE<2)
    const float* __restrict__ gsrc,  // lnn | s1 | val
    float* __restrict__ dst)         // s | v | s
{
    __shared__ float Wl[NUM_RBF * HID];
    // swizzled stage: element (k,col) -> Wl[(k>>1)*256 + col*2 + (k&1)]
    for (int t = threadIdx.x; t < NUM_RBF * HID; t += 256) {
        const int k = t >> 7, col = t & 127;
        Wl[(k >> 1) * 256 + col * 2 + (k & 1)] = Wsm[t];
    }
    __syncthreads();

    const int tid = threadIdx.x;
    const int wave = tid >> 5;
    const int lane = tid & 31;
    const int laneLo = lane & 15;
    const int laneHi = lane >> 4;
    const int ebase = blockIdx.x * 128 + wave * 16;

    // both half-waves compute the same 16 edges' geometry (lanes m and m+16)
    const int e0 = ebase + laneLo;
    const int j = jsrc[e0];
    const int i = idst[e0];
    const float px = pos[(size_t)i * 3 + 0] - pos[(size_t)j * 3 + 0];
    const float py = pos[(size_t)i * 3 + 1] - pos[(size_t)j * 3 + 1];
    const float pz = pos[(size_t)i * 3 + 2] - pos[(size_t)j * 3 + 2];
    const float d = sqrtf(px * px + py * py + pz * pz + 1e-12f);
    const float bnd = (d < CUTOFF) ? 0.5f * (cosf(d * PI_OVER_CUTOFF) + 1.0f) : 0.0f;
    const float expd = expf(-d);
    const float inv = 1.0f / (d + 1e-8f);
    const float ux = px * inv, uy = py * inv, uz = pz * inv;

    v8f acc[8];
    #pragma unroll
    for (int n = 0; n < 8; ++n) {
        const float bc = (MODE < 2) ? bias[n * 16 + laneLo] : 0.0f;
        v8f ini = {bc, bc, bc, bc, bc, bc, bc, bc};
        acc[n] = ini;
    }

    const v2f* Bp = (const v2f*)Wl;
    #pragma unroll
    for (int t3 = 0; t3 < 3; ++t3) {
        const int k0 = 4 * t3 + 2 * laneHi;
        float mu0 = RBF_M0 + (float)k0 * RBF_DM;
        float mu1 = mu0 + RBF_DM;
        float t0 = expd - mu0, t1 = expd - mu1;
        v2f a;
        a.x = bnd * expf(-RBF_BETA * t0 * t0);
        a.y = bnd * expf(-RBF_BETA * t1 * t1);
        const int prow = (2 * t3 + laneHi) * 128;
        #pragma unroll
        for (int n = 0; n < 8; ++n) {
            const v2f b = Bp[prow + n * 16 + laneLo];
            acc[n] = __builtin_amdgcn_wmma_f32_16x16x4_f32(
                false, a, false, b, (short)0, acc[n], false, false);
        }
    }

    // per-C-row edge metadata via wave32 shuffles from lanes 0..15
    int jm[8], im[8];
    float ewm[8], uxm[8], uym[8], uzm[8];
    #pragma unroll
    for (int r = 0; r < 8; ++r) {
        const int M = r + 8 * laneHi;
        jm[r] = __shfl(j, M, 32);
        im[r] = __shfl(i, M, 32);
        ewm[r] = __shfl(bnd, M, 32);
        uxm[r] = __shfl(ux, M, 32);
        uym[r] = __shfl(uy, M, 32);
        uzm[r] = __shfl(uz, M, 32);
    }

    #pragma unroll
    for (int n = 0; n < 8; ++n) {
        const int col = n * 16 + laneLo;
        #pragma unroll
        for (int r = 0; r < 8; ++r) {
            const float cv = acc[n][r];
            if (MODE == 0) {
                const float rad = silu_f(cv);
                atomicAdd(&dst[(size_t)im[r] * HID + col],
                          rad * gsrc[(size_t)jm[r] * HID + col]);
            } else if (MODE == 1) {
                const float rad = silu_f(cv);
                const float mg = rad * gsrc[(size_t)jm[r] * HID + col];
                const size_t base = (size_t)im[r] * 3 * HID + col;
                atomicAdd(&dst[base],           mg * uxm[r]);
                atomicAdd(&dst[base + HID],     mg * uym[r]);
                atomicAdd(&dst[base + 2 * HID], mg * uzm[r]);
            } else {
                const float vj = gsrc[(size_t)jm[r] * HID + col];
                atomicAdd(&dst[(size_t)im[r] * HID + col], (vj + cv) * ewm[r]);
            }
        }
    }
}

// ---------------------------------------------------------------------------
// s += sum_d v[n][d][:]^2
// ---------------------------------------------------------------------------
__global__ __launch_bounds__(256) void vsq_add_kernel(
    const float* __restrict__ v, float* __restrict__ s)
{
    const int e = blockIdx.x * 256 + threadIdx.x;   // over N*HID
    const int n = e >> 7, c = e & 127;
    const size_t b = (size_t)n * 3 * HID + c;
    const float a0 = v[b], a1 = v[b + HID], a2 = v[b + 2 * HID];
    s[e] += a0 * a0 + a1 * a1 + a2 * a2;
}

// ---------------------------------------------------------------------------
extern "C" void kernel_launch(void* const* d_in, const int* in_sizes, int n_in,
                              void* d_out, int out_size, void* d_ws, size_t ws_size,
                              hipStream_t stream)
{
    (void)in_sizes; (void)n_in; (void)out_size; (void)ws_size;

    const float* pos       = (const float*)d_in[0];
    const float* node_emb  = (const float*)d_in[1];
    const float* neigh_emb = (const float*)d_in[2];
    const float* Wr        = (const float*)d_in[3];
    const float* br        = (const float*)d_in[4];
    const float* W1        = (const float*)d_in[5];
    const float* b1        = (const float*)d_in[6];
    // d_in[7..10] = Wq, bq, Wk, bk : unused by the reference forward
    const float* Wv        = (const float*)d_in[11];
    const float* bv        = (const float*)d_in[12];
    const float* We        = (const float*)d_in[13];
    const int*   z         = (const int*)d_in[14];
    const int*   eidx      = (const int*)d_in[15];
    const int*   jsrc      = eidx;
    const int*   idst      = eidx + N_EDGES;

    float* s   = (float*)d_out;                       // [N,128] running node state
    float* lnn = (float*)d_ws;                        // [N,128]
    float* s1v = lnn + (size_t)N_NODES * HID;         // [N,128] s1 / val (reused)
    float* v   = s1v + (size_t)N_NODES * HID;         // [N,3,128]

    const int egrid = N_EDGES / 128;                  // 16 edges/wave * 8 waves
    const int ggrid = N_NODES / 64;                   // 16 rows/wave * 4 waves

    // s = node_emb[z]; lnn = LN(neigh_emb[z]); v = 0
    node_prep_kernel<<<N_NODES, 128, 0, stream>>>(z, node_emb, neigh_emb, s, lnn, v);
    // s += seg(radial * lnn[j])
    edge_pass_kernel<0><<<egrid, 256, 0, stream>>>(pos, jsrc, idst, Wr, br, lnn, s);
    // s1 = SiLU(LN(s@W1 + b1))
    gemm_ln_silu_kernel<<<ggrid, 128, 0, stream>>>(s, W1, b1, s1v);
    // v = seg(radial * unit * s1[j])
    edge_pass_kernel<1><<<egrid, 256, 0, stream>>>(pos, jsrc, idst, Wr, br, s1v, v);
    // s += sum(v*v, axis=1)
    vsq_add_kernel<<<(N_NODES * HID) / 256, 256, 0, stream>>>(v, s);

    for (int l = 0; l < NLAYERS; ++l) {
        gemm_ln_silu_kernel<<<ggrid, 128, 0, stream>>>(
            s, Wv + (size_t)l * HID * HID, bv + (size_t)l * HID, s1v);
        edge_pass_kernel<2><<<egrid, 256, 0, stream>>>(
            pos, jsrc, idst, We + (size_t)l * NUM_RBF * HID, nullptr, s1v, s);
    }
}